// GuidedFilter_4234837754115
// MI455X (gfx1250) — compile-verified
//
#include <hip/hip_runtime.h>

// Guided filter on [8,3,1024,1024] f32 via exact-f32 WMMA band-GEMM box filters.
// Box sum along an axis == multiply by banded ones matrix (clipping built in).
// P(X) = transpose(Band * X);  P(P(X)) = Box2D(X) exactly.

typedef float v2f __attribute__((ext_vector_type(2)));
typedef float v8f __attribute__((ext_vector_type(8)));

#define HW      1024
#define RAD     30
#define EPSF    1.3f
#define NPLANES 24                    // B*C = 8*3
#define PLANE   ((size_t)HW * HW)
#define TPP     ((HW / 16) * (HW / 16))   // 4096 16x16 tiles per plane

__device__ __forceinline__ v8f wmma4(v2f A, v2f B, v8f C) {
  // D = A(16x4,f32) * B(4x16,f32) + C(16x16,f32)  -- exact f32 math
  return __builtin_amdgcn_wmma_f32_16x16x4_f32(false, A, false, B, (short)0, C,
                                               false, false);
}

// clipped-window element count along one axis
__device__ __forceinline__ float cntf(int t) {
  int lo = t - RAD; if (lo < 0) lo = 0;
  int hi = t + RAD; if (hi > HW - 1) hi = HW - 1;
  return (float)(hi - lo + 1);
}

// Band operand for this lane: A[m][k], m = v0+lm (uniform band center per lane),
// k = kt + 2*lh + j, j=0,1. Value 1.0 iff |m-k|<=RAD and k < HW.
__device__ __forceinline__ v2f band_pair(int vm, int kbase) {
  v2f A;
  int k0 = kbase, k1 = kbase + 1;
  A.x = ((unsigned)(vm - k0 + RAD) <= (unsigned)(2 * RAD) && k0 < HW) ? 1.0f : 0.0f;
  A.y = ((unsigned)(vm - k1 + RAD) <= (unsigned)(2 * RAD) && k1 < HW) ? 1.0f : 0.0f;
  return A;
}

// ---------------- Pass 1: t* = P(R), P(I), P(R*I), P(R*R) ----------------
__global__ __launch_bounds__(32) void gf_pass1(
    const float* __restrict__ I, const float* __restrict__ R,
    float* __restrict__ tR, float* __restrict__ tI,
    float* __restrict__ tRI, float* __restrict__ tRR) {
  const int tile = blockIdx.x;
  const int p  = tile / TPP;
  const int t  = tile - p * TPP;
  const int u0 = (t & 63) << 4;       // input column block / output row block
  const int v0 = (t >> 6) << 4;       // band center block / output col block
  const int l  = threadIdx.x;
  const int lm = l & 15, lh = l >> 4;
  const size_t pb = (size_t)p * PLANE;
  const int vm = v0 + lm;

  v8f aR = {0,0,0,0,0,0,0,0}, aI = aR, aRI = aR, aRR = aR;

  int klo = v0 - RAD;      if (klo < 0) klo = 0;
  int khi = v0 + 15 + RAD; if (khi > HW - 1) khi = HW - 1;

  for (int kt = klo; kt <= khi; kt += 4) {
    const int kb = kt + 2 * lh;
    v2f A = band_pair(vm, kb);
    int r0 = kb     < HW - 1 ? kb     : HW - 1;   // clamp (masked by A anyway)
    int r1 = kb + 1 < HW - 1 ? kb + 1 : HW - 1;
    size_t a0 = pb + (size_t)r0 * HW + u0 + lm;
    size_t a1 = pb + (size_t)r1 * HW + u0 + lm;
    float R0 = R[a0], R1 = R[a1];
    float I0 = I[a0], I1 = I[a1];
    v2f BR  = {R0,      R1};
    v2f BI  = {I0,      I1};
    v2f BRI = {R0 * I0, R1 * I1};
    v2f BRR = {R0 * R0, R1 * R1};
    aR  = wmma4(A, BR,  aR);
    aI  = wmma4(A, BI,  aI);
    aRI = wmma4(A, BRI, aRI);
    aRR = wmma4(A, BRR, aRR);
  }

  // transposed store: lane l writes 8 contiguous floats
  const size_t ob = pb + (size_t)(u0 + lm) * HW + v0 + 8 * lh;
  *(v8f*)(tR  + ob) = aR;
  *(v8f*)(tI  + ob) = aI;
  *(v8f*)(tRI + ob) = aRI;
  *(v8f*)(tRR + ob) = aRR;
}

// ------- Pass 2: full 2-D box sums -> means -> a,b (natural layout) -------
__global__ __launch_bounds__(32) void gf_pass2(
    const float* __restrict__ tR, const float* __restrict__ tI,
    const float* __restrict__ tRI, const float* __restrict__ tRR,
    float* __restrict__ wa, float* __restrict__ wb) {
  const int tile = blockIdx.x;
  const int p  = tile / TPP;
  const int t  = tile - p * TPP;
  const int u0 = (t & 63) << 4;
  const int v0 = (t >> 6) << 4;
  const int l  = threadIdx.x;
  const int lm = l & 15, lh = l >> 4;
  const size_t pb = (size_t)p * PLANE;
  const int vm = v0 + lm;

  v8f aR = {0,0,0,0,0,0,0,0}, aI = aR, aRI = aR, aRR = aR;

  int klo = v0 - RAD;      if (klo < 0) klo = 0;
  int khi = v0 + 15 + RAD; if (khi > HW - 1) khi = HW - 1;

  for (int kt = klo; kt <= khi; kt += 4) {
    const int kb = kt + 2 * lh;
    v2f A = band_pair(vm, kb);
    int r0 = kb     < HW - 1 ? kb     : HW - 1;
    int r1 = kb + 1 < HW - 1 ? kb + 1 : HW - 1;
    size_t a0 = pb + (size_t)r0 * HW + u0 + lm;
    size_t a1 = pb + (size_t)r1 * HW + u0 + lm;
    v2f BR  = {tR[a0],  tR[a1]};
    v2f BI  = {tI[a0],  tI[a1]};
    v2f BRI = {tRI[a0], tRI[a1]};
    v2f BRR = {tRR[a0], tRR[a1]};
    aR  = wmma4(A, BR,  aR);
    aI  = wmma4(A, BI,  aI);
    aRI = wmma4(A, BRI, aRI);
    aRR = wmma4(A, BRR, aRR);
  }

  const int h = u0 + lm;          // output row (natural orientation after P∘P)
  const float cH = cntf(h);
  v8f va, vb;
#pragma unroll
  for (int i = 0; i < 8; ++i) {
    const int w = v0 + 8 * lh + i;
    const float rN  = 1.0f / (cH * cntf(w));
    const float mR  = aR[i]  * rN;
    const float mI  = aI[i]  * rN;
    const float cov = aRI[i] * rN - mR * mI;
    const float var = aRR[i] * rN - mR * mR;
    const float av  = cov / (var + EPSF);
    va[i] = av;
    vb[i] = mI - av * mR;
  }
  const size_t ob = pb + (size_t)h * HW + v0 + 8 * lh;
  *(v8f*)(wa + ob) = va;
  *(v8f*)(wb + ob) = vb;
}

// ---------------- Pass 3: ta = P(a), tb = P(b) ----------------
__global__ __launch_bounds__(32) void gf_pass3(
    const float* __restrict__ wa, const float* __restrict__ wb,
    float* __restrict__ ta, float* __restrict__ tb) {
  const int tile = blockIdx.x;
  const int p  = tile / TPP;
  const int t  = tile - p * TPP;
  const int u0 = (t & 63) << 4;
  const int v0 = (t >> 6) << 4;
  const int l  = threadIdx.x;
  const int lm = l & 15, lh = l >> 4;
  const size_t pb = (size_t)p * PLANE;
  const int vm = v0 + lm;

  v8f aA = {0,0,0,0,0,0,0,0}, aB = aA;

  int klo = v0 - RAD;      if (klo < 0) klo = 0;
  int khi = v0 + 15 + RAD; if (khi > HW - 1) khi = HW - 1;

  for (int kt = klo; kt <= khi; kt += 4) {
    const int kb = kt + 2 * lh;
    v2f A = band_pair(vm, kb);
    int r0 = kb     < HW - 1 ? kb     : HW - 1;
    int r1 = kb + 1 < HW - 1 ? kb + 1 : HW - 1;
    size_t a0 = pb + (size_t)r0 * HW + u0 + lm;
    size_t a1 = pb + (size_t)r1 * HW + u0 + lm;
    v2f BA = {wa[a0], wa[a1]};
    v2f BB = {wb[a0], wb[a1]};
    aA = wmma4(A, BA, aA);
    aB = wmma4(A, BB, aB);
  }

  const size_t ob = pb + (size_t)(u0 + lm) * HW + v0 + 8 * lh;
  *(v8f*)(ta + ob) = aA;
  *(v8f*)(tb + ob) = aB;
}

// -------- Pass 4: q = mean_a * R + mean_b  (fused final epilogue) --------
__global__ __launch_bounds__(32) void gf_pass4(
    const float* __restrict__ ta, const float* __restrict__ tb,
    const float* __restrict__ R, float* __restrict__ out) {
  const int tile = blockIdx.x;
  const int p  = tile / TPP;
  const int t  = tile - p * TPP;
  const int u0 = (t & 63) << 4;
  const int v0 = (t >> 6) << 4;
  const int l  = threadIdx.x;
  const int lm = l & 15, lh = l >> 4;
  const size_t pb = (size_t)p * PLANE;
  const int vm = v0 + lm;

  v8f aA = {0,0,0,0,0,0,0,0}, aB = aA;

  int klo = v0 - RAD;      if (klo < 0) klo = 0;
  int khi = v0 + 15 + RAD; if (khi > HW - 1) khi = HW - 1;

  for (int kt = klo; kt <= khi; kt += 4) {
    const int kb = kt + 2 * lh;
    v2f A = band_pair(vm, kb);
    int r0 = kb     < HW - 1 ? kb     : HW - 1;
    int r1 = kb + 1 < HW - 1 ? kb + 1 : HW - 1;
    size_t a0 = pb + (size_t)r0 * HW + u0 + lm;
    size_t a1 = pb + (size_t)r1 * HW + u0 + lm;
    v2f BA = {ta[a0], ta[a1]};
    v2f BB = {tb[a0], tb[a1]};
    aA = wmma4(A, BA, aA);
    aB = wmma4(A, BB, aB);
  }

  const int h = u0 + lm;
  const float cH = cntf(h);
  const size_t ob = pb + (size_t)h * HW + v0 + 8 * lh;
  const v8f rv = *(const v8f*)(R + ob);
  v8f q;
#pragma unroll
  for (int i = 0; i < 8; ++i) {
    const int w  = v0 + 8 * lh + i;
    const float rN = 1.0f / (cH * cntf(w));
    q[i] = (aA[i] * rN) * rv[i] + aB[i] * rN;
  }
  *(v8f*)(out + ob) = q;
}

extern "C" void kernel_launch(void* const* d_in, const int* in_sizes, int n_in,
                              void* d_out, int out_size, void* d_ws, size_t ws_size,
                              hipStream_t stream) {
  (void)in_sizes; (void)n_in; (void)out_size;
  const float* I = (const float*)d_in[0];
  const float* R = (const float*)d_in[1];
  float* out = (float*)d_out;
  float* ws  = (float*)d_ws;

  const size_t TOT = (size_t)NPLANES * PLANE;
  dim3 blk(32);

  if (ws_size >= 5 * TOT * sizeof(float)) {
    // Batched path: 4 launches, d_out doubles as the "b" scratch plane set.
    float* t0 = ws;
    float* t1 = ws + TOT;
    float* t2 = ws + 2 * TOT;
    float* t3 = ws + 3 * TOT;
    float* wa = ws + 4 * TOT;
    float* wb = out;
    dim3 grid((unsigned)(NPLANES * TPP));
    gf_pass1<<<grid, blk, 0, stream>>>(I, R, t0, t1, t2, t3);
    gf_pass2<<<grid, blk, 0, stream>>>(t0, t1, t2, t3, wa, wb);
    gf_pass3<<<grid, blk, 0, stream>>>(wa, wb, t0, t1);
    gf_pass4<<<grid, blk, 0, stream>>>(t0, t1, R, out);
  } else {
    // Per-plane fallback: 5 plane-sized scratch buffers (20 MB).
    float* t0 = ws;
    float* t1 = ws + PLANE;
    float* t2 = ws + 2 * PLANE;
    float* t3 = ws + 3 * PLANE;
    float* wa = ws + 4 * PLANE;
    dim3 grid((unsigned)TPP);
    for (int p = 0; p < NPLANES; ++p) {
      const float* Ip = I + (size_t)p * PLANE;
      const float* Rp = R + (size_t)p * PLANE;
      float* op = out + (size_t)p * PLANE;
      gf_pass1<<<grid, blk, 0, stream>>>(Ip, Rp, t0, t1, t2, t3);
      gf_pass2<<<grid, blk, 0, stream>>>(t0, t1, t2, t3, wa, op);
      gf_pass3<<<grid, blk, 0, stream>>>(wa, op, t0, t1);
      gf_pass4<<<grid, blk, 0, stream>>>(t0, t1, Rp, op);
    }
  }
}